// SolveNewsvendor_58394375356724
// MI455X (gfx1250) — compile-verified
//
#include <hip/hip_runtime.h>

// Batched PDIP newsvendor QP, exploiting the arrowhead structure of
// diag(q) + G' D G  (G = [[-1,-I,0],[1,0,-I],[-I_N]]).  One wave32 per batch
// item; each lane owns 4 of the K=128 quantile indices.  Wave-wide sum
// reductions are performed with V_WMMA_F32_16X16X4_F32 (B = all-ones, so the
// result is exact f32 and independent of the B-matrix striping).

namespace {
constexpr int   KQ    = 128;       // quantiles
constexpr int   NB    = 128;       // batch
constexpr int   U     = KQ / 32;   // indices per lane (wave32)
constexpr int   NITER = 25;
constexpr float SIGMA = 0.1f;
constexpr float BIG   = 1e9f;
}

typedef __attribute__((ext_vector_type(2))) float v2f;
typedef __attribute__((ext_vector_type(8))) float v8f;

// 32-lane sum, broadcast to all lanes.
// A VGPR0 = v, VGPR1 = 0  =>  A(m,0)=v_m, A(m,2)=v_{m+16}, A(m,1)=A(m,3)=0.
// B = all ones (layout-independent) => D(m,j) = v_m + v_{m+16}.
// Lane l holds D(r + 8*(l>>4), l&15) in VGPR r; summing its 8 D VGPRs gives
// one half of the total, and one xor-16 shuffle completes the reduction.
__device__ __forceinline__ float wave_sum(float v) {
#if __has_builtin(__builtin_amdgcn_wmma_f32_16x16x4_f32)
  v2f a;  a[0]  = v;    a[1]  = 0.0f;
  v2f bo; bo[0] = 1.0f; bo[1] = 1.0f;
  v8f c = {};
  v8f d = __builtin_amdgcn_wmma_f32_16x16x4_f32(
      /*neg_a=*/false, a, /*neg_b=*/false, bo,
      /*c_mod=*/(short)0, c, /*reuse_a=*/false, /*reuse_b=*/false);
  float s = ((d[0] + d[1]) + (d[2] + d[3])) + ((d[4] + d[5]) + (d[6] + d[7]));
  s += __shfl_xor(s, 16, 32);
  return s;
#else
  #pragma unroll
  for (int m = 16; m >= 1; m >>= 1) v += __shfl_xor(v, m, 32);
  return v;
#endif
}

__device__ __forceinline__ float wave_min(float v) {
  #pragma unroll
  for (int m = 16; m >= 1; m >>= 1) v = fminf(v, __shfl_xor(v, m, 32));
  return v;
}

__global__ void __launch_bounds__(32)
newsvendor_pdip(const float* __restrict__ y,   // (B,K)
                const float* __restrict__ Qd,  // (N,)
                const float* __restrict__ pv,  // (N,)
                const float* __restrict__ h,   // (M,)  h[0..K-1] = -d
                float* __restrict__ out) {     // (B,)
  const int b    = blockIdx.x;
  const int lane = threadIdx.x;                // 0..31

  const float q0 = Qd[0];
  const float p0 = pv[0];

  float qb[U], qh[U], pbv[U], phv[U], dd[U];
  #pragma unroll
  for (int u = 0; u < U; ++u) {
    const int   i  = lane + 32 * u;
    const float yv = y[b * KQ + i];
    dd[u]  = -h[i];                  // = d_i
    qb[u]  = Qd[1 + i]      * yv;
    qh[u]  = Qd[1 + KQ + i] * yv;
    pbv[u] = pv[1 + i]      * yv;
    phv[u] = pv[1 + KQ + i] * yv;
  }

  // State: z = (z0, zb, zh);  s,lam over M = [blk1(K), blk2(K), blk3(N=2K+1)]
  float zb[U], zh[U];
  float s1[U], s2[U], s3b[U], s3h[U];
  float l1[U], l2[U], l3b[U], l3h[U];
  float z0 = 0.0f, s30 = 1.0f, l30 = 1.0f;
  #pragma unroll
  for (int u = 0; u < U; ++u) {
    zb[u] = zh[u] = 0.0f;
    s1[u] = s2[u] = s3b[u] = s3h[u] = 1.0f;
    l1[u] = l2[u] = l3b[u] = l3h[u] = 1.0f;
  }

  const float Minv = 1.0f / (float)(3 * KQ + 1);

  #pragma unroll 1
  for (int it = 0; it < NITER; ++it) {
    // ---- Phase A: state-only quantities + 4 sum reductions -------------
    float dinv1[U], dinv2[U], dinv3b[U], dinv3h[U], Db[U], Dh[U];
    float pa1 = 0.f, pa2 = 0.f, pa3 = 0.f, pa4 = 0.f;
    #pragma unroll
    for (int u = 0; u < U; ++u) {
      dinv1[u]  = l1[u]  / s1[u];
      dinv2[u]  = l2[u]  / s2[u];
      dinv3b[u] = l3b[u] / s3b[u];
      dinv3h[u] = l3h[u] / s3h[u];
      Db[u] = qb[u] + dinv1[u] + dinv3b[u];   // arrowhead diagonal (buy)
      Dh[u] = qh[u] + dinv2[u] + dinv3h[u];   // arrowhead diagonal (hold)
      pa1 += s1[u]*l1[u] + s2[u]*l2[u] + s3b[u]*l3b[u] + s3h[u]*l3h[u];
      pa2 += l2[u] - l1[u];
      pa3 += dinv1[u] + dinv2[u];
      pa4 += dinv1[u]*dinv1[u]/Db[u] + dinv2[u]*dinv2[u]/Dh[u];
    }
    const float A1 = wave_sum(pa1);
    const float A2 = wave_sum(pa2);
    const float A3 = wave_sum(pa3);
    const float A4 = wave_sum(pa4);

    const float dinv30 = l30 / s30;
    const float mu   = (A1 + s30 * l30) * Minv;           // sum(s*lam)/(3K+1)
    const float smu  = SIGMA * mu;
    const float denom = (q0 + A3 + dinv30) - A4;          // Schur denominator
    const float rd0  = q0 * z0 + p0 + A2 - l30;           // r_dual[0]
    const float rp30 = -z0 + s30;                         // r_prim (blk3, j=0)
    const float w30  = dinv30 * rp30 + smu / s30 - l30;

    // ---- Phase B: residuals, rhs, 2 sum reductions ---------------------
    float rp1[U], rp2[U], rp3b[U], rp3h[U];
    float w1[U], w2[U], w3b[U], w3h[U], rhsb[U], rhsh[U];
    float pb1 = 0.f, pb3 = 0.f;
    #pragma unroll
    for (int u = 0; u < U; ++u) {
      rp1[u]  = -z0 - zb[u] + s1[u] + dd[u];
      rp2[u]  =  z0 - zh[u] + s2[u] - dd[u];
      rp3b[u] = -zb[u] + s3b[u];
      rp3h[u] = -zh[u] + s3h[u];
      w1[u]  = dinv1[u]*rp1[u]   + smu/s1[u]  - l1[u];
      w2[u]  = dinv2[u]*rp2[u]   + smu/s2[u]  - l2[u];
      w3b[u] = dinv3b[u]*rp3b[u] + smu/s3b[u] - l3b[u];
      w3h[u] = dinv3h[u]*rp3h[u] + smu/s3h[u] - l3h[u];
      const float rdb = qb[u]*zb[u] + pbv[u] - l1[u] - l3b[u];
      const float rdh = qh[u]*zh[u] + phv[u] - l2[u] - l3h[u];
      rhsb[u] = w1[u] + w3b[u] - rdb;     // rhs = -(r_dual + w@G)
      rhsh[u] = w2[u] + w3h[u] - rdh;
      pb1 += w2[u] - w1[u];
      pb3 += dinv1[u]*rhsb[u]/Db[u] - dinv2[u]*rhsh[u]/Dh[u];
    }
    const float B1 = wave_sum(pb1);
    const float B3 = wave_sum(pb3);

    // ---- Arrowhead (Schur) solve --------------------------------------
    const float rhs0 = -(rd0 + B1 - w30);
    const float dz0  = (rhs0 - B3) / denom;

    const float g30  = -dz0;              // (G dz) blk3 j=0
    const float ds30 = -rp30 - g30;
    const float dl30 = dinv30 * g30 + w30;
    float pcs = (ds30 < 0.f) ? (-s30 / ds30) : BIG;
    float pcl = (dl30 < 0.f) ? (-l30 / dl30) : BIG;

    float dzb[U], dzh[U];
    float ds1[U], ds2[U], ds3b[U], ds3h[U];
    float dl1[U], dl2[U], dl3b[U], dl3h[U];
    #pragma unroll
    for (int u = 0; u < U; ++u) {
      dzb[u] = (rhsb[u] - dinv1[u]*dz0) / Db[u];
      dzh[u] = (rhsh[u] + dinv2[u]*dz0) / Dh[u];
      const float g1  = -dz0 - dzb[u];
      const float g2  =  dz0 - dzh[u];
      const float g3b = -dzb[u];
      const float g3h = -dzh[u];
      ds1[u]  = -rp1[u]  - g1;
      ds2[u]  = -rp2[u]  - g2;
      ds3b[u] = -rp3b[u] - g3b;
      ds3h[u] = -rp3h[u] - g3h;
      dl1[u]  = dinv1[u]*g1   + w1[u];    // dlam = dinv*(Gdz+rp)+smu/s-lam
      dl2[u]  = dinv2[u]*g2   + w2[u];
      dl3b[u] = dinv3b[u]*g3b + w3b[u];
      dl3h[u] = dinv3h[u]*g3h + w3h[u];
      pcs = fminf(pcs, (ds1[u]  < 0.f) ? (-s1[u]  / ds1[u])  : BIG);
      pcs = fminf(pcs, (ds2[u]  < 0.f) ? (-s2[u]  / ds2[u])  : BIG);
      pcs = fminf(pcs, (ds3b[u] < 0.f) ? (-s3b[u] / ds3b[u]) : BIG);
      pcs = fminf(pcs, (ds3h[u] < 0.f) ? (-s3h[u] / ds3h[u]) : BIG);
      pcl = fminf(pcl, (dl1[u]  < 0.f) ? (-l1[u]  / dl1[u])  : BIG);
      pcl = fminf(pcl, (dl2[u]  < 0.f) ? (-l2[u]  / dl2[u])  : BIG);
      pcl = fminf(pcl, (dl3b[u] < 0.f) ? (-l3b[u] / dl3b[u]) : BIG);
      pcl = fminf(pcl, (dl3h[u] < 0.f) ? (-l3h[u] / dl3h[u]) : BIG);
    }
    const float a_s   = wave_min(pcs);
    const float a_l   = wave_min(pcl);
    const float alpha = fminf(1.0f, 0.99f * fminf(a_s, a_l));

    z0  += alpha * dz0;
    s30 += alpha * ds30;
    l30 += alpha * dl30;
    #pragma unroll
    for (int u = 0; u < U; ++u) {
      zb[u]  += alpha * dzb[u];
      zh[u]  += alpha * dzh[u];
      s1[u]  += alpha * ds1[u];
      s2[u]  += alpha * ds2[u];
      s3b[u] += alpha * ds3b[u];
      s3h[u] += alpha * ds3h[u];
      l1[u]  += alpha * dl1[u];
      l2[u]  += alpha * dl2[u];
      l3b[u] += alpha * dl3b[u];
      l3h[u] += alpha * dl3h[u];
    }
  }

  if (lane == 0) out[b] = z0;   // reference returns z[:, :1]
}

extern "C" void kernel_launch(void* const* d_in, const int* in_sizes, int n_in,
                              void* d_out, int out_size, void* d_ws, size_t ws_size,
                              hipStream_t stream) {
  (void)in_sizes; (void)n_in; (void)d_ws; (void)ws_size; (void)out_size;
  const float* y  = (const float*)d_in[0];   // (B,K)
  const float* Qd = (const float*)d_in[1];   // (N,)
  const float* pv = (const float*)d_in[2];   // (N,)
  // d_in[3] = G : its exact block structure is hard-coded above
  const float* h  = (const float*)d_in[4];   // (M,)
  newsvendor_pdip<<<dim3(NB), dim3(32), 0, stream>>>(y, Qd, pv, h, (float*)d_out);
}